// MessagePassingElectionModel_17686675325014
// MI455X (gfx1250) — compile-verified
//
#include <hip/hip_runtime.h>
#include <hip/hip_bf16.h>
#include <math.h>

typedef __attribute__((ext_vector_type(2))) float v2f;
typedef __attribute__((ext_vector_type(8))) float v8f;

#define WMMA_F32(a, b, c) \
  __builtin_amdgcn_wmma_f32_16x16x4_f32(false, (a), false, (b), (short)0, (c), false, false)

__device__ __forceinline__ void atomicAddF(float* p, float v) {
  __hip_atomic_fetch_add(p, v, __ATOMIC_RELAXED, __HIP_MEMORY_SCOPE_AGENT);
}

// order-preserving uint encoding of float for integer atomicMax
__device__ __forceinline__ unsigned encf(float x) {
  unsigned u = __float_as_uint(x);
  return (u & 0x80000000u) ? ~u : (u | 0x80000000u);
}
__device__ __forceinline__ float decf(unsigned u) {
  u = (u & 0x80000000u) ? (u & 0x7fffffffu) : ~u;
  return __uint_as_float(u);
}

// ---------------------------------------------------------------- h init
__global__ void __launch_bounds__(256)
k_init_h(const float* __restrict__ x, const float* __restrict__ wi,
         const float* __restrict__ bi, float* __restrict__ h, int N) {
  int total = N * 32;
  for (int i = blockIdx.x * blockDim.x + threadIdx.x; i < total;
       i += gridDim.x * blockDim.x) {
    int n = i >> 5, d = i & 31;
    h[i] = fmaf(x[n * 2 + 0], wi[d * 2 + 0], fmaf(x[n * 2 + 1], wi[d * 2 + 1], bi[d]));
  }
}

// ---------------------------------------------------------------- zero
__global__ void __launch_bounds__(256)
k_zero_u32(unsigned* __restrict__ p, int n) {
  for (int i = blockIdx.x * blockDim.x + threadIdx.x; i < n;
       i += gridDim.x * blockDim.x)
    p[i] = 0u;
}

// ---------------------------------------------------------------- gemm1
// t1[e][n] = sum_k msg[e][k] * w1[n][k]   (msg = [h[dst] | h[src] | ea])
// per 16-edge tile, per wave: 16 K-slices x 2 N-tiles of V_WMMA_F32_16X16X4_F32
__global__ void __launch_bounds__(256)
k_gemm1(const float* __restrict__ h, const int* __restrict__ src,
        const int* __restrict__ dst, const float* __restrict__ ea,
        const float* __restrict__ w1l,  // [32][65] for this layer
        float* __restrict__ t, float* __restrict__ stats, int E) {
  const int lane = threadIdx.x & 31;
  const int l16 = lane & 15;
  const int half = lane >> 4;
  const int nwaves = (gridDim.x * blockDim.x) >> 5;
  const int wid = (blockIdx.x * blockDim.x + threadIdx.x) >> 5;
  const int ntiles = E >> 4;

  // preload B operands: B[k][n] = w1[n][k]
  v2f Bd[8][2], Bs[8][2];
#pragma unroll
  for (int s = 0; s < 8; ++s) {
    const int k = 4 * s + 2 * half;
#pragma unroll
    for (int nt = 0; nt < 2; ++nt) {
      const float* wr = w1l + (nt * 16 + l16) * 65;
      v2f bd; bd.x = wr[k];      bd.y = wr[k + 1];      Bd[s][nt] = bd;
      v2f bs; bs.x = wr[32 + k]; bs.y = wr[32 + k + 1]; Bs[s][nt] = bs;
    }
  }
  const float wea0 = w1l[l16 * 65 + 64];
  const float wea1 = w1l[(16 + l16) * 65 + 64];

  float s0 = 0.f, q0 = 0.f, s1 = 0.f, q1 = 0.f;

  for (int tile = wid; tile < ntiles; tile += nwaves) {
    const int e0 = tile << 4;
    const int id = dst[e0 + l16];
    const int is = src[e0 + l16];
    const float* hd = h + (size_t)id * 32 + 2 * half;
    const float* hs = h + (size_t)is * 32 + 2 * half;
    const float eav = ea[e0 + l16];

    v8f c0 = {}; v8f c1 = {};
#pragma unroll
    for (int s = 0; s < 8; ++s) {  // K = 0..31  (h[dst])
      v2f a = *(const v2f*)(hd + 4 * s);
      c0 = WMMA_F32(a, Bd[s][0], c0);
      c1 = WMMA_F32(a, Bd[s][1], c1);
    }
#pragma unroll
    for (int s = 0; s < 8; ++s) {  // K = 32..63 (h[src])
      v2f a = *(const v2f*)(hs + 4 * s);
      c0 = WMMA_F32(a, Bs[s][0], c0);
      c1 = WMMA_F32(a, Bs[s][1], c1);
    }
    // rank-1 edge_attr term (K = 64), store, and BN stats
    float* trow = t + (size_t)e0 * 32;
#pragma unroll
    for (int r = 0; r < 8; ++r) {
      const int m = 8 * half + r;
      const float eam = __shfl(eav, m, 32);
      const float v0 = c0[r] + eam * wea0;
      const float v1 = c1[r] + eam * wea1;
      trow[m * 32 + l16] = v0;
      trow[m * 32 + 16 + l16] = v1;
      s0 += v0; q0 += v0 * v0;
      s1 += v1; q1 += v1 * v1;
    }
  }
  s0 += __shfl_xor(s0, 16, 32); q0 += __shfl_xor(q0, 16, 32);
  s1 += __shfl_xor(s1, 16, 32); q1 += __shfl_xor(q1, 16, 32);
  if (lane < 16) {
    atomicAddF(&stats[l16], s0);       atomicAddF(&stats[32 + l16], q0);
    atomicAddF(&stats[16 + l16], s1);  atomicAddF(&stats[48 + l16], q1);
  }
}

// ---------------------------------------------------------------- BN reduce
// ss[n] = scale, ss[32+n] = shift so that normalized = t*scale + shift
__global__ void k_bnreduce(const float* __restrict__ stats,
                           const float* __restrict__ g, const float* __restrict__ b,
                           float* __restrict__ ss, float invE) {
  const int n = threadIdx.x;  // 32 threads
  const float mu = stats[n] * invE;
  const float var = stats[32 + n] * invE - mu * mu;
  const float a = rsqrtf(var + 1e-5f) * g[n];
  ss[n] = a;
  ss[32 + n] = b[n] - mu * a;
}

// ---------------------------------------------------------------- gemm2 (in place)
// m1 = relu(t1*scale1+shift1); t2[e][n] = sum_k m1[e][k]*w2[n][k]; overwrite t
__global__ void __launch_bounds__(256)
k_gemm2(float* __restrict__ t, const float* __restrict__ w2l,
        const float* __restrict__ ss1, float* __restrict__ stats, int E) {
  const int lane = threadIdx.x & 31;
  const int l16 = lane & 15;
  const int half = lane >> 4;
  const int nwaves = (gridDim.x * blockDim.x) >> 5;
  const int wid = (blockIdx.x * blockDim.x + threadIdx.x) >> 5;
  const int ntiles = E >> 4;

  v2f B[8][2];
  float sc[8][2], sh[8][2];
#pragma unroll
  for (int s = 0; s < 8; ++s) {
    const int k = 4 * s + 2 * half;
#pragma unroll
    for (int nt = 0; nt < 2; ++nt) {
      const float* wr = w2l + (nt * 16 + l16) * 32;
      v2f bb; bb.x = wr[k]; bb.y = wr[k + 1]; B[s][nt] = bb;
    }
    sc[s][0] = ss1[k];     sh[s][0] = ss1[32 + k];
    sc[s][1] = ss1[k + 1]; sh[s][1] = ss1[32 + k + 1];
  }

  float s0 = 0.f, q0 = 0.f, s1 = 0.f, q1 = 0.f;

  for (int tile = wid; tile < ntiles; tile += nwaves) {
    float* trow = t + (size_t)(tile << 4) * 32;
    v8f c0 = {}; v8f c1 = {};
#pragma unroll
    for (int s = 0; s < 8; ++s) {
      v2f a = *(const v2f*)(trow + l16 * 32 + 4 * s + 2 * half);
      a.x = fmaxf(0.f, fmaf(a.x, sc[s][0], sh[s][0]));
      a.y = fmaxf(0.f, fmaf(a.y, sc[s][1], sh[s][1]));
      c0 = WMMA_F32(a, B[s][0], c0);
      c1 = WMMA_F32(a, B[s][1], c1);
    }
#pragma unroll
    for (int r = 0; r < 8; ++r) {
      const int m = 8 * half + r;
      const float v0 = c0[r], v1 = c1[r];
      trow[m * 32 + l16] = v0;
      trow[m * 32 + 16 + l16] = v1;
      s0 += v0; q0 += v0 * v0;
      s1 += v1; q1 += v1 * v1;
    }
  }
  s0 += __shfl_xor(s0, 16, 32); q0 += __shfl_xor(q0, 16, 32);
  s1 += __shfl_xor(s1, 16, 32); q1 += __shfl_xor(q1, 16, 32);
  if (lane < 16) {
    atomicAddF(&stats[l16], s0);       atomicAddF(&stats[32 + l16], q0);
    atomicAddF(&stats[16 + l16], s1);  atomicAddF(&stats[48 + l16], q1);
  }
}

// ---------------------------------------------------------------- scatter
// h[dst[e]][n] += relu(t2[e][n]*scale2+shift2)
__global__ void __launch_bounds__(256)
k_scatter(const float* __restrict__ t, const float* __restrict__ ss2,
          const int* __restrict__ dst, float* __restrict__ h, int E) {
  const long long total = (long long)E * 32;
  for (long long i = blockIdx.x * (long long)blockDim.x + threadIdx.x; i < total;
       i += (long long)gridDim.x * blockDim.x) {
    const int n = (int)(i & 31);
    const int e = (int)(i >> 5);
    float v = fmaxf(0.f, fmaf(t[i], ss2[n], ss2[32 + n]));
    atomicAddF(&h[(size_t)dst[e] * 32 + n], v);
  }
}

// ---------------------------------------------------------------- logits
__global__ void __launch_bounds__(256)
k_logits(const float* __restrict__ h, const int* __restrict__ cand,
         const float* __restrict__ low, const float* __restrict__ lob,
         float* __restrict__ logits, int C) {
  const int lane = threadIdx.x & 31;
  const int nwaves = (gridDim.x * blockDim.x) >> 5;
  const int wid = (blockIdx.x * blockDim.x + threadIdx.x) >> 5;
  for (int c = wid; c < C; c += nwaves) {
    float v = h[(size_t)cand[c] * 32 + lane] * low[lane];
#pragma unroll
    for (int off = 16; off >= 1; off >>= 1) v += __shfl_xor(v, off, 32);
    if (lane == 0) logits[c] = v + lob[0];
  }
}

// ---------------------------------------------------------------- segmented softmax
__global__ void __launch_bounds__(256)
k_segmax(const float* __restrict__ logits, const int* __restrict__ cand,
         const int* __restrict__ batch, unsigned* __restrict__ maxenc, int C) {
  for (int i = blockIdx.x * blockDim.x + threadIdx.x; i < C;
       i += gridDim.x * blockDim.x)
    atomicMax(&maxenc[batch[cand[i]]], encf(logits[i]));
}

__global__ void __launch_bounds__(256)
k_segsum(const float* __restrict__ logits, const int* __restrict__ cand,
         const int* __restrict__ batch, const unsigned* __restrict__ maxenc,
         float* __restrict__ sumexp, int C) {
  for (int i = blockIdx.x * blockDim.x + threadIdx.x; i < C;
       i += gridDim.x * blockDim.x) {
    const int s = batch[cand[i]];
    atomicAddF(&sumexp[s], expf(logits[i] - decf(maxenc[s])));
  }
}

__global__ void __launch_bounds__(256)
k_final(const float* __restrict__ logits, const int* __restrict__ cand,
        const int* __restrict__ batch, const unsigned* __restrict__ maxenc,
        const float* __restrict__ sumexp, float* __restrict__ out, int C) {
  for (int i = blockIdx.x * blockDim.x + threadIdx.x; i < C;
       i += gridDim.x * blockDim.x) {
    const int s = batch[cand[i]];
    out[i] = logits[i] - (decf(maxenc[s]) + logf(sumexp[s]));
  }
}

// ---------------------------------------------------------------- launch
extern "C" void kernel_launch(void* const* d_in, const int* in_sizes, int n_in,
                              void* d_out, int out_size, void* d_ws, size_t ws_size,
                              hipStream_t stream) {
  const float* x     = (const float*)d_in[0];
  const int*   eidx  = (const int*)d_in[1];
  const float* ea    = (const float*)d_in[2];
  const int*   cand  = (const int*)d_in[3];
  const int*   batch = (const int*)d_in[4];
  const float* wi    = (const float*)d_in[5];
  const float* bi    = (const float*)d_in[6];
  const float* w1    = (const float*)d_in[7];
  const float* g1    = (const float*)d_in[9];
  const float* bb1   = (const float*)d_in[10];
  const float* w2    = (const float*)d_in[11];
  const float* g2    = (const float*)d_in[13];
  const float* bb2   = (const float*)d_in[14];
  const float* low   = (const float*)d_in[15];
  const float* lob   = (const float*)d_in[16];
  float* out = (float*)d_out;

  const int N = in_sizes[0] / 2;
  const int E = in_sizes[1] / 2;
  const int C = in_sizes[3];
  const int L = in_sizes[7] / (32 * 65);
  const int G = 1000;  // NUM_GRAPHS

  const int* src = eidx;       // edge_index[0]
  const int* dst = eidx + E;   // edge_index[1]

  float* ws     = (float*)d_ws;
  float* h      = ws;                          // [N,32]
  float* t      = h + (size_t)N * 32;          // [E,32] (reused for t1/t2)
  float* small  = t + (size_t)E * 32;
  float* stats1 = small;                       // 64
  float* stats2 = small + 64;                  // 64
  float* ss1    = small + 128;                 // 64 (scale|shift)
  float* ss2    = small + 192;                 // 64
  float* logits = small + 256;                 // C
  unsigned* maxenc = (unsigned*)(logits + C);  // G
  float* sumexp = (float*)(maxenc + G);        // G

  const float invE = 1.0f / (float)E;

  k_init_h<<<2048, 256, 0, stream>>>(x, wi, bi, h, N);

  for (int l = 0; l < L; ++l) {
    k_zero_u32<<<1, 128, 0, stream>>>((unsigned*)stats1, 128);  // stats1+stats2
    k_gemm1<<<1024, 256, 0, stream>>>(h, src, dst, ea, w1 + (size_t)l * 32 * 65,
                                      t, stats1, E);
    k_bnreduce<<<1, 32, 0, stream>>>(stats1, g1 + l * 32, bb1 + l * 32, ss1, invE);
    k_gemm2<<<1024, 256, 0, stream>>>(t, w2 + (size_t)l * 32 * 32, ss1, stats2, E);
    k_bnreduce<<<1, 32, 0, stream>>>(stats2, g2 + l * 32, bb2 + l * 32, ss2, invE);
    k_scatter<<<2048, 256, 0, stream>>>(t, ss2, dst, h, E);
  }

  k_logits<<<160, 256, 0, stream>>>(h, cand, low, lob, logits, C);
  k_zero_u32<<<8, 256, 0, stream>>>(maxenc, 2 * G);  // maxenc + sumexp
  k_segmax<<<40, 256, 0, stream>>>(logits, cand, batch, maxenc, C);
  k_segsum<<<40, 256, 0, stream>>>(logits, cand, batch, maxenc, sumexp, C);
  k_final<<<40, 256, 0, stream>>>(logits, cand, batch, maxenc, sumexp, out, C);
}